// RungeKuttaIntegratorCell_46669114638375
// MI455X (gfx1250) — compile-verified
//
#include <hip/hip_runtime.h>
#include <hip/hip_bf16.h>

// ---------------------------------------------------------------------------
// Batched LTI scan:  x_{t+1} = Ad x_t + Bd u_t ,  y_t = x_t[0:2]
// B=4096 batches, T=4096 steps, state=4, input=2.  fp32 throughout.
//
// Chunked time-parallel formulation (Ad constant):
//   x_{t+CH} = A^CH x_t + sum_k A^{CH-1-k} B u_{t+k}
// Chunk contributions computed with V_WMMA_F32_16X16X4_F32 (fp32 matrix pipe).
// ---------------------------------------------------------------------------

typedef float v2f __attribute__((ext_vector_type(2)));
typedef float v8f __attribute__((ext_vector_type(8)));

#define BATCH  4096
#define TSTEPS 4096
#define CH     128
#define NC     (TSTEPS / CH)   /* 32 chunks per batch */
#define KSTEPS ((2 * CH) / 4)  /* 64 WMMA K-steps per chunk (K=4 each) */

// workspace layout (float offsets)
#define OFF_AD    0                          /* Ad 4x4 row-major      (16) */
#define OFF_BD    16                         /* Bd 4x2 row-major      (8)  */
#define OFF_ACH   24                         /* A^CH 4x4 row-major    (16) */
#define OFF_AEB   64                         /* A^e*B, e=0..CH-1, [e][n][c] (CH*8) */
#define OFF_PB    (OFF_AEB + CH * 8)         /* pre-swizzled WMMA B operand: [kk][lane][2] */
#define OFF_CAGG  (OFF_PB + KSTEPS * 64)     /* chunk contributions [b][j][4] */
#define OFF_XIN   (OFF_CAGG + BATCH * NC * 4)/* chunk-initial states [b][j][4] */
#define WS_FLOATS (OFF_XIN + BATCH * NC * 4)
#define WS_BYTES  ((size_t)WS_FLOATS * 4)    /* ~4.2 MB */

// ---------------------------------------------------------------------------
// Discretize: Ad,Bd = expm(DT * [[Ac,Bc],[0,0]])[:4]  (double Taylor; ||X||~1.2
// so 26 terms converge far past fp32 precision)
// ---------------------------------------------------------------------------
__device__ void discretize_f64(const float* cp, double Ad[4][4], double Bd[4][2]) {
    const double M0 = 20.0, M1 = 10.0;
    const double K0 = 2000.0, K1 = 1000.0, K2 = 5000.0;
    const double dt = 0.002;
    double c1 = (double)cp[0], c2 = (double)cp[1], c3 = (double)cp[2];

    double em[6][6];
    for (int i = 0; i < 6; ++i)
        for (int j = 0; j < 6; ++j) em[i][j] = 0.0;
    em[0][2] = 1.0; em[1][3] = 1.0;
    em[2][0] = -(K0 + K1) / M0;  em[2][1] =  K1 / M0;
    em[3][0] =  K1 / M1;         em[3][1] = -(K1 + K2) / M1;
    em[2][2] = -(c1 + c2) / M0;  em[2][3] =  c2 / M0;
    em[3][2] =  c2 / M1;         em[3][3] = -(c2 + c3) / M1;
    em[2][4] = 1.0 / M0;         em[3][5] = 1.0 / M1;
    for (int i = 0; i < 6; ++i)
        for (int j = 0; j < 6; ++j) em[i][j] *= dt;

    double S[6][6], Tm[6][6], tmp[6][6];
    for (int i = 0; i < 6; ++i)
        for (int j = 0; j < 6; ++j) { S[i][j] = (i == j) ? 1.0 : 0.0; Tm[i][j] = S[i][j]; }
    for (int it = 1; it <= 26; ++it) {
        for (int r = 0; r < 6; ++r)
            for (int c = 0; c < 6; ++c) {
                double s = 0.0;
                for (int k = 0; k < 6; ++k) s += Tm[r][k] * em[k][c];
                tmp[r][c] = s / (double)it;
            }
        for (int r = 0; r < 6; ++r)
            for (int c = 0; c < 6; ++c) { Tm[r][c] = tmp[r][c]; S[r][c] += tmp[r][c]; }
    }
    for (int r = 0; r < 4; ++r) {
        for (int c = 0; c < 4; ++c) Ad[r][c] = S[r][c];
        for (int c = 0; c < 2; ++c) Bd[r][c] = S[r][4 + c];
    }
}

// ---------------------------------------------------------------------------
// Setup: Ad/Bd/A^CH/A^e*B, then pre-swizzle PB into the 4x16 fp32 WMMA
// B-matrix layout (VGPR0: K=0 lanes0-15 / K=2 lanes16-31; VGPR1: K=1 / K=3).
// ---------------------------------------------------------------------------
__global__ void setup_kernel(const float* __restrict__ cp, float* __restrict__ ws) {
    if (threadIdx.x == 0) {
        double Ad[4][4], Bd[4][2];
        discretize_f64(cp, Ad, Bd);
        for (int r = 0; r < 4; ++r) {
            for (int c = 0; c < 4; ++c) ws[OFF_AD + r * 4 + c] = (float)Ad[r][c];
            for (int c = 0; c < 2; ++c) ws[OFF_BD + r * 2 + c] = (float)Bd[r][c];
        }
        // A^e * B for e = 0..CH-1
        double W[4][2], Wn[4][2];
        for (int n = 0; n < 4; ++n)
            for (int c = 0; c < 2; ++c) W[n][c] = Bd[n][c];
        for (int e = 0; e < CH; ++e) {
            for (int n = 0; n < 4; ++n)
                for (int c = 0; c < 2; ++c) ws[OFF_AEB + e * 8 + n * 2 + c] = (float)W[n][c];
            for (int n = 0; n < 4; ++n)
                for (int c = 0; c < 2; ++c) {
                    double s = 0.0;
                    for (int k = 0; k < 4; ++k) s += Ad[n][k] * W[k][c];
                    Wn[n][c] = s;
                }
            for (int n = 0; n < 4; ++n)
                for (int c = 0; c < 2; ++c) W[n][c] = Wn[n][c];
        }
        // A^CH
        double X[4][4], Xn[4][4];
        for (int r = 0; r < 4; ++r)
            for (int c = 0; c < 4; ++c) X[r][c] = (r == c) ? 1.0 : 0.0;
        for (int e = 0; e < CH; ++e) {
            for (int r = 0; r < 4; ++r)
                for (int c = 0; c < 4; ++c) {
                    double s = 0.0;
                    for (int k = 0; k < 4; ++k) s += Ad[r][k] * X[k][c];
                    Xn[r][c] = s;
                }
            for (int r = 0; r < 4; ++r)
                for (int c = 0; c < 4; ++c) X[r][c] = Xn[r][c];
        }
        for (int r = 0; r < 4; ++r)
            for (int c = 0; c < 4; ++c) ws[OFF_ACH + r * 4 + c] = (float)X[r][c];
    }
    __syncthreads();   // global writes by thread 0 visible to block

    int kk = threadIdx.x;           // one kk per thread, KSTEPS = blockDim
    if (kk < KSTEPS) {
        for (int lane = 0; lane < 32; ++lane) {
            int n = lane & 15, hi = lane >> 4;
            for (int r = 0; r < 2; ++r) {
                int q = kk * 4 + hi * 2 + r;       // flattened K index = 2k + c
                int k = q >> 1, c = q & 1;
                int e = CH - 1 - k;
                float v = (n < 4) ? ws[OFF_AEB + e * 8 + n * 2 + c] : 0.0f;
                ws[OFF_PB + (kk * 32 + lane) * 2 + r] = v;
            }
        }
    }
}

// ---------------------------------------------------------------------------
// Pass 1: chunk contribution Cagg[b][j][:] = sum_k A^{CH-1-k} B u[b][jCH+k]
// One wave per (16-batch, chunk) tile; 64 accumulating fp32 WMMAs.
// A operand: 2 consecutive u floats/lane per K-step (16x4 fp32 A layout).
// ---------------------------------------------------------------------------
__global__ void __launch_bounds__(32)
pass1_kernel(const float* __restrict__ u, const float* __restrict__ ws,
             float* __restrict__ cagg) {
    int tile = blockIdx.x;
    int j  = tile % NC;
    int bb = (tile / NC) * 16;
    int lane = threadIdx.x;
    int m = lane & 15, half = lane >> 4;

    const float* PB = ws + OFF_PB;
    size_t ubase = ((size_t)(bb + m) * TSTEPS + (size_t)j * CH) * 2 + (size_t)half * 2;

    v8f acc = {};
    for (int kk = 0; kk < KSTEPS; ++kk) {
        v2f a = *(const v2f*)(u + ubase + (size_t)kk * 4);
        v2f b = *(const v2f*)(PB + (size_t)(kk * 32 + lane) * 2);
        acc = __builtin_amdgcn_wmma_f32_16x16x4_f32(
            /*neg_a=*/false, a, /*neg_b=*/false, b,
            /*c_mod=*/(short)0, acc, /*reuse_a=*/false, /*reuse_b=*/false);
    }
    // D layout: VGPR v holds M=v (lanes 0-15) / M=v+8 (lanes 16-31), N=lane%16
    if (m < 4) {
#pragma unroll
        for (int v = 0; v < 8; ++v) {
            int M = v + half * 8;  // batch offset within tile
            cagg[(((size_t)(bb + M) * NC) + (size_t)j) * 4 + m] = acc[v];
        }
    }
}

// ---------------------------------------------------------------------------
// Inter-chunk scan: x_{j+1} = A^CH x_j + Cagg[j]; record chunk-initial states.
// ---------------------------------------------------------------------------
__global__ void scan_kernel(const float* __restrict__ x0, float* __restrict__ ws) {
    int b = blockIdx.x * blockDim.x + threadIdx.x;
    if (b >= BATCH) return;
    float A[16];
#pragma unroll
    for (int i = 0; i < 16; ++i) A[i] = ws[OFF_ACH + i];
    float x[4];
#pragma unroll
    for (int i = 0; i < 4; ++i) x[i] = x0[(size_t)b * 4 + i];

    const float* cg = ws + OFF_CAGG + (size_t)b * NC * 4;
    float*       xi = ws + OFF_XIN  + (size_t)b * NC * 4;
    for (int j = 0; j < NC; ++j) {
#pragma unroll
        for (int i = 0; i < 4; ++i) xi[j * 4 + i] = x[i];
        float nx[4];
#pragma unroll
        for (int r = 0; r < 4; ++r)
            nx[r] = fmaf(A[r * 4 + 0], x[0],
                    fmaf(A[r * 4 + 1], x[1],
                    fmaf(A[r * 4 + 2], x[2],
                    fmaf(A[r * 4 + 3], x[3], cg[j * 4 + r]))));
#pragma unroll
        for (int i = 0; i < 4; ++i) x[i] = nx[i];
    }
}

// ---------------------------------------------------------------------------
// Pass 2: replay each chunk from its initial state, streaming y (bulk BW).
// One thread per (batch, chunk): 131072 threads.
// ---------------------------------------------------------------------------
__global__ void pass2_kernel(const float* __restrict__ u, const float* __restrict__ ws,
                             float* __restrict__ y) {
    size_t gid = (size_t)blockIdx.x * blockDim.x + threadIdx.x;
    if (gid >= (size_t)BATCH * NC) return;
    int b = (int)(gid / NC), j = (int)(gid % NC);

    float A[16], Bf[8];
#pragma unroll
    for (int i = 0; i < 16; ++i) A[i] = ws[OFF_AD + i];
#pragma unroll
    for (int i = 0; i < 8; ++i) Bf[i] = ws[OFF_BD + i];

    float x[4];
#pragma unroll
    for (int i = 0; i < 4; ++i) x[i] = ws[OFF_XIN + ((size_t)b * NC + j) * 4 + i];

    size_t base = ((size_t)b * TSTEPS + (size_t)j * CH) * 2;
    const float* up = u + base;
    float*       yp = y + base;

    for (int k = 0; k < CH; ++k) {
        yp[2 * k]     = x[0];
        yp[2 * k + 1] = x[1];
        float u0 = up[2 * k], u1 = up[2 * k + 1];
        float nx[4];
#pragma unroll
        for (int r = 0; r < 4; ++r)
            nx[r] = fmaf(A[r * 4 + 0], x[0],
                    fmaf(A[r * 4 + 1], x[1],
                    fmaf(A[r * 4 + 2], x[2],
                    fmaf(A[r * 4 + 3], x[3],
                    fmaf(Bf[r * 2 + 0], u0,
                         Bf[r * 2 + 1] * u1)))));
#pragma unroll
        for (int i = 0; i < 4; ++i) x[i] = nx[i];
    }
}

// ---------------------------------------------------------------------------
// Fallback (ws too small): correct thread-per-batch sequential scan.
// ---------------------------------------------------------------------------
__global__ void naive_kernel(const float* __restrict__ u, const float* __restrict__ x0,
                             const float* __restrict__ cp, float* __restrict__ y) {
    int b = blockIdx.x * blockDim.x + threadIdx.x;
    if (b >= BATCH) return;
    double Ad[4][4], Bd[4][2];
    discretize_f64(cp, Ad, Bd);
    float A[16], Bf[8];
    for (int r = 0; r < 4; ++r) {
        for (int c = 0; c < 4; ++c) A[r * 4 + c] = (float)Ad[r][c];
        for (int c = 0; c < 2; ++c) Bf[r * 2 + c] = (float)Bd[r][c];
    }
    float x[4];
#pragma unroll
    for (int i = 0; i < 4; ++i) x[i] = x0[(size_t)b * 4 + i];
    const float* up = u + (size_t)b * TSTEPS * 2;
    float*       yp = y + (size_t)b * TSTEPS * 2;
    for (int t = 0; t < TSTEPS; ++t) {
        yp[2 * t] = x[0]; yp[2 * t + 1] = x[1];
        float u0 = up[2 * t], u1 = up[2 * t + 1];
        float nx[4];
#pragma unroll
        for (int r = 0; r < 4; ++r)
            nx[r] = fmaf(A[r * 4 + 0], x[0],
                    fmaf(A[r * 4 + 1], x[1],
                    fmaf(A[r * 4 + 2], x[2],
                    fmaf(A[r * 4 + 3], x[3],
                    fmaf(Bf[r * 2 + 0], u0, Bf[r * 2 + 1] * u1)))));
#pragma unroll
        for (int i = 0; i < 4; ++i) x[i] = nx[i];
    }
}

extern "C" void kernel_launch(void* const* d_in, const int* in_sizes, int n_in,
                              void* d_out, int out_size, void* d_ws, size_t ws_size,
                              hipStream_t stream) {
    (void)in_sizes; (void)n_in; (void)out_size;
    const float* u  = (const float*)d_in[0];   // (B, T, 2)
    const float* x0 = (const float*)d_in[1];   // (B, 4)
    const float* cp = (const float*)d_in[2];   // (3,)
    float* y  = (float*)d_out;                 // (B, T, 2)
    float* ws = (float*)d_ws;

    if (ws_size >= WS_BYTES) {
        setup_kernel<<<1, KSTEPS, 0, stream>>>(cp, ws);
        pass1_kernel<<<(BATCH / 16) * NC, 32, 0, stream>>>(u, ws, ws + OFF_CAGG);
        scan_kernel<<<BATCH / 256, 256, 0, stream>>>(x0, ws);
        int p2threads = BATCH * NC;
        pass2_kernel<<<(p2threads + 255) / 256, 256, 0, stream>>>(u, ws, y);
    } else {
        naive_kernel<<<BATCH / 256, 256, 0, stream>>>(u, x0, cp, y);
    }
}